// Factorization_73280732005209
// MI455X (gfx1250) — compile-verified
//
#include <hip/hip_runtime.h>

typedef __attribute__((ext_vector_type(2))) float v2f;
typedef __attribute__((ext_vector_type(8))) float v8f;

#define BB 256
#define MM 512
#define NN 256
#define KK 16

// One workgroup per batch b. 256 threads = 8 waves (wave32).
// Phase 1: xv[b] = x[b] @ v via v_wmma_f32_16x16x4_f32 (two independent
//          accumulator chains to keep the matrix pipe busy), l = x[b] @ w on VALU.
// Phase 2: sv = column-sum of xv (LDS reduction).
// Phase 3: out = l + bias + 0.5*(xv.sv - ||xv||^2).
__global__ __launch_bounds__(256) void fm_fwd_kernel(
    const float* __restrict__ x,      // [B, M, N]
    const float* __restrict__ w,      // [N]
    const float* __restrict__ bias,   // [1]
    const float* __restrict__ v,      // [N, K]
    float* __restrict__ out)          // [B, M]
{
    __shared__ v2f   vstage[64 * 32];     // [chunk][lane] -> (v[4c+2h][col], v[4c+2h+1][col])
    __shared__ float wlds[NN];
    __shared__ float xv[MM * 20];         // stride 20: rows r, r+8 hit disjoint LDS bank groups
    __shared__ float lrow[MM];
    __shared__ float svpart[16][16];
    __shared__ float sv[KK];

    const int tid  = threadIdx.x;
    const int lane = tid & 31;
    const int wave = tid >> 5;
    const int half = lane >> 4;   // 0: K pair {0,1}; 1: K pair {2,3}
    const int col  = lane & 15;
    const int b    = blockIdx.x;

    // --- stage w and pre-swizzled B fragments of v ---
    wlds[tid] = w[tid];                       // blockDim == NN == 256
    for (int idx = tid; idx < 64 * 32; idx += 256) {
        const int c  = idx >> 5;
        const int l2 = idx & 31;
        const int h  = l2 >> 4;
        const int cc = l2 & 15;
        const int r0 = 4 * c + 2 * h;         // v row
        v2f t;
        t.x = v[r0 * KK + cc];
        t.y = v[(r0 + 1) * KK + cc];
        vstage[idx] = t;
    }
    __syncthreads();

    const float* xb = x + (size_t)b * (MM * NN);

    // --- phase 1: GEMM xv = x[b] @ v  (+ linear term on the side) ---
    for (int t = 0; t < 4; ++t) {
        const int mbase = wave * 64 + t * 16;
        // A-fragment (16x4 f32): lane<16 holds row M=lane, K={0,1}; lane>=16 holds K={2,3}
        const float* arow = xb + (size_t)(mbase + col) * NN + 2 * half;

        // prefetch next tile's stream (same lane pattern, +16 rows)
        if (t < 3) {
            __builtin_prefetch(arow + 16 * NN, 0, 1);
            __builtin_prefetch(arow + 16 * NN + 128, 0, 1);
        }

        v8f acc0 = {0.f, 0.f, 0.f, 0.f, 0.f, 0.f, 0.f, 0.f};
        v8f acc1 = {0.f, 0.f, 0.f, 0.f, 0.f, 0.f, 0.f, 0.f};
        float lacc = 0.f;
        #pragma unroll 4
        for (int c = 0; c < 64; c += 2) {
            const v2f a0  = *(const v2f*)(arow + 4 * c);            // global_load_b64
            const v2f a1  = *(const v2f*)(arow + 4 * (c + 1));
            const v2f bf0 = vstage[c * 32 + lane];                  // ds_load_b64
            const v2f bf1 = vstage[(c + 1) * 32 + lane];
            const v2f wv0 = *(const v2f*)(&wlds[4 * c + 2 * half]);
            const v2f wv1 = *(const v2f*)(&wlds[4 * (c + 1) + 2 * half]);
            acc0 = __builtin_amdgcn_wmma_f32_16x16x4_f32(
                false, a0, false, bf0, (short)0, acc0, false, false);
            acc1 = __builtin_amdgcn_wmma_f32_16x16x4_f32(
                false, a1, false, bf1, (short)0, acc1, false, false);
            lacc += a0.x * wv0.x + a0.y * wv0.y;
            lacc += a1.x * wv1.x + a1.y * wv1.y;
        }
        const v8f acc = acc0 + acc1;

        // C/D layout: VGPR r, lane<16 -> (M=r, N=col); lane>=16 -> (M=r+8, N=col)
        #pragma unroll
        for (int r = 0; r < 8; ++r) {
            const int row = mbase + r + 8 * half;
            xv[row * 20 + col] = acc[r];
        }
        // rows' linear partials live in lane m and lane m+16
        lacc += __shfl_xor(lacc, 16, 32);
        if (half == 0) lrow[mbase + col] = lacc;
    }
    __syncthreads();

    // --- phase 2: sv[k] = sum_m xv[m][k] ---
    {
        const int k = tid & 15;
        const int g = tid >> 4;
        float s = 0.f;
        for (int m = g; m < MM; m += 16) s += xv[m * 20 + k];
        svpart[g][k] = s;
    }
    __syncthreads();
    if (tid < 16) {
        float s = 0.f;
        #pragma unroll
        for (int g = 0; g < 16; ++g) s += svpart[g][tid];
        sv[tid] = s;
    }
    __syncthreads();

    // --- phase 3: out[b,m] = l + bias + 0.5*(xv.sv - ||xv||^2) ---
    const float bsc = bias[0];
    float* outb = out + (size_t)b * MM;
    for (int m = tid; m < MM; m += 256) {
        float rs = 0.f, dg = 0.f;
        #pragma unroll
        for (int k = 0; k < KK; ++k) {
            const float q = xv[m * 20 + k];
            rs += q * sv[k];
            dg += q * q;
        }
        outb[m] = lrow[m] + bsc + 0.5f * (rs - dg);
    }
}

extern "C" void kernel_launch(void* const* d_in, const int* in_sizes, int n_in,
                              void* d_out, int out_size, void* d_ws, size_t ws_size,
                              hipStream_t stream) {
    const float* x    = (const float*)d_in[0];   // [B, M, N]
    const float* w    = (const float*)d_in[1];   // [1, N]
    const float* bias = (const float*)d_in[2];   // [1]
    const float* v    = (const float*)d_in[3];   // [N, K]
    float* out = (float*)d_out;                  // [B, M, 1]
    (void)in_sizes; (void)n_in; (void)out_size; (void)d_ws; (void)ws_size;

    fm_fwd_kernel<<<dim3(BB), dim3(256), 0, stream>>>(x, w, bias, v, out);
}